// TGNN_59124519796826
// MI455X (gfx1250) — compile-verified
//
#include <hip/hip_runtime.h>
#include <hip/hip_bf16.h>
#include <math.h>

// ---------------------------------------------------------------------------
// CDNA5 (gfx1250) fp32 WMMA fragments: V_WMMA_F32_16X16X4_F32
//   A (16x4 f32): lane<16 -> row=lane, {A[r][k], A[r][k+1]}; lane>=16 -> {A[r][k+2], A[r][k+3]}
//   B (4x16 f32): lane<16 -> col=lane, {B[k][c], B[k+1][c]}; lane>=16 -> {B[k+2][c], B[k+3][c]}
//   C/D (16x16):  vgpr j: lane<16 -> row j, lane>=16 -> row j+8; col = lane&15
// ---------------------------------------------------------------------------
typedef __attribute__((ext_vector_type(2))) float v2f;
typedef __attribute__((ext_vector_type(8))) float v8f;

__device__ __forceinline__ v8f wmma_f32x4(v2f a, v2f b, v8f c) {
    return __builtin_amdgcn_wmma_f32_16x16x4_f32(
        /*neg_a=*/false, a, /*neg_b=*/false, b,
        /*c_mod=*/(short)0, c, /*reuse_a=*/false, /*reuse_b=*/false);
}

// Generic -> LDS byte offset (AS3 ptrtoint): matches async-to-LDS VDST semantics
__device__ __forceinline__ unsigned lds_off(void* p) {
    return (unsigned)(unsigned long long)(__attribute__((address_space(3))) void*)p;
}

// Async 16B/lane copy global -> LDS (ASYNCcnt-tracked, no VGPR round trip)
__device__ __forceinline__ void async_ld_b128(unsigned lds_byte_off, const void* gptr) {
    asm volatile("global_load_async_to_lds_b128 %0, %1, off"
                 :: "v"(lds_byte_off), "v"(gptr) : "memory");
}
__device__ __forceinline__ void wait_async0() {
    asm volatile("s_wait_asynccnt 0" ::: "memory");
}

__device__ __forceinline__ float atomicMaxFloat(float* addr, float val) {
    unsigned int* ua = (unsigned int*)addr;
    unsigned int old = *ua;
    while (__uint_as_float(old) < val) {
        unsigned int assumed = old;
        old = atomicCAS(ua, assumed, __float_as_uint(val));
        if (old == assumed) break;
    }
    return __uint_as_float(old);
}

// ---------------------------------------------------------------------------
// Gather x = memory_table[n_id], lu = last_update_table[n_id]
// ---------------------------------------------------------------------------
__global__ void gather_kernel(const float* __restrict__ mem, const float* __restrict__ lut,
                              const int* __restrict__ n_id,
                              float* __restrict__ x, float* __restrict__ lu, int N_) {
    int tid = blockIdx.x * blockDim.x + threadIdx.x;
    if (tid >= N_ * 128) return;
    int i = tid >> 7, c = tid & 127;
    int node = n_id[i];
    x[tid] = mem[(size_t)node * 128 + c];
    if (c == 0) lu[i] = lut[node];
}

// ---------------------------------------------------------------------------
// Dense fp32 WMMA GEMM: out[nrows x NC] = A[nrows x K] @ W[K x NC] + bias
// W staged per-block into LDS, fragment-swizzled (conflict-free ds_load_b64).
// ---------------------------------------------------------------------------
template <int K, int NC>
__global__ void node_gemm_kernel(const float* __restrict__ A, const float* __restrict__ W,
                                 const float* __restrict__ bias, float* __restrict__ out,
                                 int nrows) {
    constexpr int NT = NC / 16;
    constexpr int NSTEP = K / 4;
    __shared__ v2f LWs[NSTEP * NT * 32];       // K*NC*4 bytes (<= 64 KB)

    const int lane = threadIdx.x & 31;
    const int wave = threadIdx.x >> 5;
    const int nw   = (int)(blockDim.x >> 5);
    const int hl   = lane >> 4;
    const int lr   = lane & 15;

    for (int f = wave; f < NSTEP * NT; f += nw) {
        int s = f / NT, t = f - s * NT;
        const float* wp = W + (size_t)(s * 4 + hl * 2) * NC + t * 16 + lr;
        v2f p; p.x = wp[0]; p.y = wp[NC];
        LWs[f * 32 + lane] = p;
    }
    __syncthreads();

    const int r0 = (blockIdx.x * nw + wave) * 16;
    if (r0 >= nrows) return;

    v8f acc[NT] = {};
    const float* ap = A + (size_t)(r0 + lr) * K + hl * 2;
#pragma unroll 4
    for (int s = 0; s < NSTEP; ++s) {
        v2f a; a.x = ap[s * 4]; a.y = ap[s * 4 + 1];
#pragma unroll
        for (int t = 0; t < NT; ++t) {
            v2f b = LWs[(s * NT + t) * 32 + lane];
            acc[t] = wmma_f32x4(a, b, acc[t]);
        }
    }
#pragma unroll
    for (int t = 0; t < NT; ++t) {
        int col = t * 16 + lr;
        float bb = bias ? bias[col] : 0.0f;
#pragma unroll
        for (int j = 0; j < 8; ++j) {
            int row = r0 + j + hl * 8;
            out[(size_t)row * NC + col] = acc[t][j] + bb;
        }
    }
}

// ---------------------------------------------------------------------------
// Fused 4-weight GEMM sharing one A stream (Q/K/V/skip):
// A fragments cached in registers (read x exactly once); one 64 KB LDS stage
// reused across the four weight matrices with barrier-bracketed restaging.
// ---------------------------------------------------------------------------
template <int K, int NC>
__global__ void node_gemm4_kernel(const float* __restrict__ A,
                                  const float* __restrict__ Wm0, const float* __restrict__ bb0, float* __restrict__ oo0,
                                  const float* __restrict__ Wm1, const float* __restrict__ bb1, float* __restrict__ oo1,
                                  const float* __restrict__ Wm2, const float* __restrict__ bb2, float* __restrict__ oo2,
                                  const float* __restrict__ Wm3, const float* __restrict__ bb3, float* __restrict__ oo3,
                                  int nrows) {
    constexpr int NT = NC / 16;
    constexpr int NSTEP = K / 4;
    __shared__ v2f LWs[NSTEP * NT * 32];       // 64 KB

    const int lane = threadIdx.x & 31;
    const int wave = threadIdx.x >> 5;
    const int nw   = (int)(blockDim.x >> 5);
    const int hl   = lane >> 4;
    const int lr   = lane & 15;

    const int r0 = (blockIdx.x * nw + wave) * 16;
    const bool active = (r0 < nrows);           // wave-uniform -> EXEC all-1s in WMMA

    v2f areg[NSTEP];
    if (active) {
        const float* ap = A + (size_t)(r0 + lr) * K + hl * 2;
#pragma unroll 8
        for (int s = 0; s < NSTEP; ++s) { v2f a; a.x = ap[s * 4]; a.y = ap[s * 4 + 1]; areg[s] = a; }
    }
    const float* Ws[4] = {Wm0, Wm1, Wm2, Wm3};
    const float* bs[4] = {bb0, bb1, bb2, bb3};
    float*       os[4] = {oo0, oo1, oo2, oo3};

    for (int m = 0; m < 4; ++m) {
        __syncthreads();                        // previous W fully consumed
        for (int f = wave; f < NSTEP * NT; f += nw) {
            int s = f / NT, t = f - s * NT;
            const float* wp = Ws[m] + (size_t)(s * 4 + hl * 2) * NC + t * 16 + lr;
            v2f p; p.x = wp[0]; p.y = wp[NC];
            LWs[f * 32 + lane] = p;
        }
        __syncthreads();
        if (active) {
            v8f acc[NT] = {};
#pragma unroll 4
            for (int s = 0; s < NSTEP; ++s) {
#pragma unroll
                for (int t = 0; t < NT; ++t)
                    acc[t] = wmma_f32x4(areg[s], LWs[(s * NT + t) * 32 + lane], acc[t]);
            }
            const float* bias = bs[m];
            float* out = os[m];
#pragma unroll
            for (int t = 0; t < NT; ++t) {
                int col = t * 16 + lr;
                float bb = bias[col];
#pragma unroll
                for (int j = 0; j < 8; ++j)
                    out[(size_t)(r0 + j + hl * 8) * NC + col] = acc[t][j] + bb;
            }
        }
    }
}

// ---------------------------------------------------------------------------
// Edge pass 1: edge_attr -> e_proj (WMMA) -> logits + per-dst atomic max
// We staged in DYNAMIC LDS (64 KB, fragment-swizzled) shared by the block;
// msg half of the tile arrives via global_load_async_to_lds_b128 overlapping
// the cos() time-encoding.
// ---------------------------------------------------------------------------
#define EW 4
__global__ void edge_pass1(const float* __restrict__ lu, const float* __restrict__ graph_t,
                           const float* __restrict__ graph_msg,
                           const float* __restrict__ Wt, const float* __restrict__ bt,
                           const float* __restrict__ We,
                           const float* __restrict__ q, const float* __restrict__ k,
                           const int* __restrict__ esrc, const int* __restrict__ edst,
                           const int* __restrict__ e_id,
                           float* __restrict__ logits, float* __restrict__ mmax, int E_) {
    __shared__ __align__(16) float tile[EW][16][132];
    extern __shared__ v2f WeS[];               // 32*8*32 v2f = 64 KB (dynamic)
    const int lane = threadIdx.x & 31;
    const int wave = threadIdx.x >> 5;
    const int hl = lane >> 4, lr = lane & 15;

    // Cooperative swizzled staging of We (all threads, before the edge guard)
    for (int f = wave; f < 32 * 8; f += EW) {
        int s = f >> 3, t = f & 7;
        const float* wp = We + (size_t)(s * 4 + hl * 2) * 128 + t * 16 + lr;
        v2f p; p.x = wp[0]; p.y = wp[128];
        WeS[f * 32 + lane] = p;
    }
    __syncthreads();

    const int e0 = (blockIdx.x * EW + wave) * 16;
    if (e0 >= E_) return;
    float (*T)[132] = tile[wave];

    // Async copy first 16 floats of msg rows -> tile cols 64..79 (16 rows x 64B = 2 x 512B)
#pragma unroll
    for (int i = 0; i < 2; ++i) {
        int g = i * 32 + lane;
        int ei = g >> 2, ch = g & 3;
        int eid = e_id[e0 + ei];
        async_ld_b128(lds_off(&T[ei][64 + ch * 4]),
                      graph_msg + (size_t)eid * 64 + ch * 4);
    }
    // cos time-encoding -> tile cols 0..63 (overlaps the async copies)
    for (int ei = 0; ei < 16; ++ei) {
        int e = e0 + ei;
        float rel = lu[esrc[e]] - graph_t[e_id[e]];
        int c = lane * 2;
        T[ei][c]     = cosf(rel * Wt[c] + bt[c]);
        T[ei][c + 1] = cosf(rel * Wt[c + 1] + bt[c + 1]);
    }
    wait_async0();

    // e_proj = edge_attr @ We (B from LDS: one ds_load_b64 per WMMA)
    v8f acc[8] = {};
#pragma unroll 2
    for (int s = 0; s < 32; ++s) {
        v2f a; a.x = T[lr][s * 4 + hl * 2]; a.y = T[lr][s * 4 + hl * 2 + 1];
#pragma unroll
        for (int t = 0; t < 8; ++t)
            acc[t] = wmma_f32x4(a, WeS[(s * 8 + t) * 32 + lane], acc[t]);
    }
    // Stage e_proj back to LDS (DS ops of one wave are in-order)
#pragma unroll
    for (int t = 0; t < 8; ++t)
#pragma unroll
        for (int j = 0; j < 8; ++j)
            T[j + hl * 8][t * 16 + lr] = acc[t][j];

    // logits: lane -> (edge = lane&15, head = lane>>4); b128 gathers
    {
        int ei = lane & 15, h = lane >> 4;
        int e = e0 + ei;
        int se = esrc[e], de = edst[e];
        const float4* qd = (const float4*)(q + (size_t)de * 128 + h * 64);
        const float4* ks = (const float4*)(k + (size_t)se * 128 + h * 64);
        const float4* tp = (const float4*)(&T[ei][h * 64]);
        float s = 0.0f;
#pragma unroll 4
        for (int d4 = 0; d4 < 16; ++d4) {
            float4 qv = qd[d4], kv = ks[d4], tv = tp[d4];
            s += qv.x * (kv.x + tv.x) + qv.y * (kv.y + tv.y)
               + qv.z * (kv.z + tv.z) + qv.w * (kv.w + tv.w);
        }
        s *= 0.125f;  // 1/sqrt(D=64)
        logits[(size_t)e * 2 + h] = s;
        atomicMaxFloat(&mmax[(size_t)de * 2 + h], s);
    }
}

// ---------------------------------------------------------------------------
// Edge pass 2: w = exp(logit - m[dst]); s[dst] += w
// ---------------------------------------------------------------------------
__global__ void edge_pass2(const float* __restrict__ logits, const float* __restrict__ mmax,
                           const int* __restrict__ edst,
                           float* __restrict__ wbuf, float* __restrict__ sbuf, int E_) {
    int tid = blockIdx.x * blockDim.x + threadIdx.x;
    if (tid >= E_ * 2) return;
    int e = tid >> 1, h = tid & 1;
    int de = edst[e];
    float w = expf(logits[tid] - mmax[(size_t)de * 2 + h]);
    wbuf[tid] = w;
    atomicAdd(&sbuf[(size_t)de * 2 + h], w);
}

// ---------------------------------------------------------------------------
// Edge pass 3: recompute e_proj (WMMA), agg[dst] += alpha * (v[src] + e_proj)
// ---------------------------------------------------------------------------
__global__ void edge_pass3(const float* __restrict__ lu, const float* __restrict__ graph_t,
                           const float* __restrict__ graph_msg,
                           const float* __restrict__ Wt, const float* __restrict__ bt,
                           const float* __restrict__ We,
                           const float* __restrict__ v,
                           const float* __restrict__ wbuf, const float* __restrict__ sbuf,
                           const int* __restrict__ esrc, const int* __restrict__ edst,
                           const int* __restrict__ e_id,
                           float* __restrict__ agg, int E_) {
    __shared__ __align__(16) float tile[EW][16][132];
    extern __shared__ v2f WeS[];               // 64 KB dynamic
    const int lane = threadIdx.x & 31;
    const int wave = threadIdx.x >> 5;
    const int hl = lane >> 4, lr = lane & 15;

    for (int f = wave; f < 32 * 8; f += EW) {
        int s = f >> 3, t = f & 7;
        const float* wp = We + (size_t)(s * 4 + hl * 2) * 128 + t * 16 + lr;
        v2f p; p.x = wp[0]; p.y = wp[128];
        WeS[f * 32 + lane] = p;
    }
    __syncthreads();

    const int e0 = (blockIdx.x * EW + wave) * 16;
    if (e0 >= E_) return;
    float (*T)[132] = tile[wave];

#pragma unroll
    for (int i = 0; i < 2; ++i) {
        int g = i * 32 + lane;
        int ei = g >> 2, ch = g & 3;
        int eid = e_id[e0 + ei];
        async_ld_b128(lds_off(&T[ei][64 + ch * 4]),
                      graph_msg + (size_t)eid * 64 + ch * 4);
    }
    for (int ei = 0; ei < 16; ++ei) {
        int e = e0 + ei;
        float rel = lu[esrc[e]] - graph_t[e_id[e]];
        int c = lane * 2;
        T[ei][c]     = cosf(rel * Wt[c] + bt[c]);
        T[ei][c + 1] = cosf(rel * Wt[c + 1] + bt[c + 1]);
    }
    wait_async0();

    v8f acc[8] = {};
#pragma unroll 2
    for (int s = 0; s < 32; ++s) {
        v2f a; a.x = T[lr][s * 4 + hl * 2]; a.y = T[lr][s * 4 + hl * 2 + 1];
#pragma unroll
        for (int t = 0; t < 8; ++t)
            acc[t] = wmma_f32x4(a, WeS[(s * 8 + t) * 32 + lane], acc[t]);
    }
#pragma unroll
    for (int t = 0; t < 8; ++t)
#pragma unroll
        for (int j = 0; j < 8; ++j)
            T[j + hl * 8][t * 16 + lr] = acc[t][j];

    for (int ei = 0; ei < 16; ++ei) {
        int e = e0 + ei;
        int se = esrc[e], de = edst[e];
        int c0 = lane * 4;
        int h = c0 >> 6;
        float alpha = wbuf[(size_t)e * 2 + h] / (sbuf[(size_t)de * 2 + h] + 1e-16f);
        float4 vv = *(const float4*)(v + (size_t)se * 128 + c0);
        float4 tv = *(const float4*)(&T[ei][c0]);
        float* ag = agg + (size_t)de * 128 + c0;
        atomicAdd(ag + 0, alpha * (vv.x + tv.x));
        atomicAdd(ag + 1, alpha * (vv.y + tv.y));
        atomicAdd(ag + 2, alpha * (vv.z + tv.z));
        atomicAdd(ag + 3, alpha * (vv.w + tv.w));
    }
}

// ---------------------------------------------------------------------------
__global__ void add_skip_kernel(float* __restrict__ agg, const float* __restrict__ skip, int n) {
    int tid = blockIdx.x * blockDim.x + threadIdx.x;
    if (tid < n) agg[tid] += skip[tid];
}

__global__ void bn_reduce(const float* __restrict__ h1, float* __restrict__ sums,
                          float* __restrict__ sqs, int N_) {
    int c = threadIdx.x;  // 128 threads
    float s = 0.0f, ss = 0.0f;
    for (int r = blockIdx.x; r < N_; r += gridDim.x) {
        float v = h1[(size_t)r * 128 + c];
        s += v; ss += v * v;
    }
    atomicAdd(&sums[c], s);
    atomicAdd(&sqs[c], ss);
}

__global__ void bn_apply(float* __restrict__ h1, const float* __restrict__ sums,
                         const float* __restrict__ sqs, const float* __restrict__ gamma,
                         const float* __restrict__ beta, int N_) {
    int tid = blockIdx.x * blockDim.x + threadIdx.x;
    if (tid >= N_ * 128) return;
    int c = tid & 127;
    float inv_n = 1.0f / (float)N_;
    float mu = sums[c] * inv_n;
    float var = sqs[c] * inv_n - mu * mu;
    float v = (h1[tid] - mu) * rsqrtf(var + 1e-5f) * gamma[c] + beta[c];
    h1[tid] = fmaxf(v, 0.0f);
}

// ---------------------------------------------------------------------------
// Link predictor: out = relu(z2[ia] @ Wa + ba + z2[ib] @ Wb + bb) @ Wf + bf
// ---------------------------------------------------------------------------
#define LW 4
__global__ void linkpred_kernel(const float* __restrict__ z2,
                                const int* __restrict__ ia, const int* __restrict__ ib,
                                const float* __restrict__ Wa, const float* __restrict__ ba,
                                const float* __restrict__ Wb, const float* __restrict__ bb,
                                const float* __restrict__ Wf, const float* __restrict__ bf,
                                float* __restrict__ out, int B_) {
    __shared__ __align__(16) float A1[LW][16][68];
    __shared__ __align__(16) float A2[LW][16][68];
    const int lane = threadIdx.x & 31;
    const int wave = threadIdx.x >> 5;
    const int hl = lane >> 4, lr = lane & 15;
    const int r0 = (blockIdx.x * LW + wave) * 16;
    if (r0 >= B_) return;
    float (*T1)[68] = A1[wave];
    float (*T2)[68] = A2[wave];

#pragma unroll
    for (int i = 0; i < 8; ++i) {
        int g = i * 32 + lane;
        int ei = g >> 4, ch = g & 15;
        int ra = ia[r0 + ei];
        int rb = ib[r0 + ei];
        async_ld_b128(lds_off(&T1[ei][ch * 4]), z2 + (size_t)ra * 64 + ch * 4);
        async_ld_b128(lds_off(&T2[ei][ch * 4]), z2 + (size_t)rb * 64 + ch * 4);
    }
    wait_async0();

    v8f acc[4] = {};
#pragma unroll 4
    for (int kk = 0; kk < 64; kk += 4) {
        v2f a1; a1.x = T1[lr][kk + hl * 2]; a1.y = T1[lr][kk + hl * 2 + 1];
        v2f a2; a2.x = T2[lr][kk + hl * 2]; a2.y = T2[lr][kk + hl * 2 + 1];
        const float* wpa = Wa + (size_t)(kk + hl * 2) * 64 + lr;
        const float* wpb = Wb + (size_t)(kk + hl * 2) * 64 + lr;
#pragma unroll
        for (int t = 0; t < 4; ++t) {
            v2f b1; b1.x = wpa[t * 16]; b1.y = wpa[64 + t * 16];
            acc[t] = wmma_f32x4(a1, b1, acc[t]);
            v2f b2; b2.x = wpb[t * 16]; b2.y = wpb[64 + t * 16];
            acc[t] = wmma_f32x4(a2, b2, acc[t]);
        }
    }
#pragma unroll
    for (int t = 0; t < 4; ++t) {
        int col = t * 16 + lr;
        float bias = ba[col] + bb[col];
#pragma unroll
        for (int j = 0; j < 8; ++j)
            T1[j + hl * 8][col] = fmaxf(acc[t][j] + bias, 0.0f);
    }
    if (lane < 16) {
        float s = 0.0f;
        for (int c = 0; c < 64; ++c) s += T1[lane][c] * Wf[c];
        out[r0 + lane] = s + bf[0];
    }
}

__global__ void fill_neg_inf(float* __restrict__ p, int n) {
    int tid = blockIdx.x * blockDim.x + threadIdx.x;
    if (tid < n) p[tid] = -3.0e38f;
}

// ---------------------------------------------------------------------------
// Host side
// ---------------------------------------------------------------------------
extern "C" void kernel_launch(void* const* d_in, const int* in_sizes, int n_in,
                              void* d_out, int out_size, void* d_ws, size_t ws_size,
                              hipStream_t stream) {
    const float* memory_table = (const float*)d_in[0];
    const float* last_update  = (const float*)d_in[1];
    const float* graph_t      = (const float*)d_in[2];
    const float* graph_msg    = (const float*)d_in[3];
    const float* Wt  = (const float*)d_in[4];
    const float* bt  = (const float*)d_in[5];
    const float* Wq  = (const float*)d_in[6];
    const float* bq  = (const float*)d_in[7];
    const float* Wk  = (const float*)d_in[8];
    const float* bk  = (const float*)d_in[9];
    const float* Wv  = (const float*)d_in[10];
    const float* bv  = (const float*)d_in[11];
    const float* We  = (const float*)d_in[12];
    const float* Wsk = (const float*)d_in[13];
    const float* bsk = (const float*)d_in[14];
    const float* W1  = (const float*)d_in[15];
    const float* b1  = (const float*)d_in[16];
    const float* bng = (const float*)d_in[17];
    const float* bnb = (const float*)d_in[18];
    const float* W2  = (const float*)d_in[19];
    const float* b2  = (const float*)d_in[20];
    const float* Wls = (const float*)d_in[21];
    const float* bls = (const float*)d_in[22];
    const float* Wld = (const float*)d_in[23];
    const float* bld = (const float*)d_in[24];
    const float* Wlf = (const float*)d_in[25];
    const float* blf = (const float*)d_in[26];
    const int* n_id  = (const int*)d_in[27];
    const int* eidx  = (const int*)d_in[28];
    const int* e_id  = (const int*)d_in[29];
    const int* s_idx = (const int*)d_in[30];
    const int* d_idx = (const int*)d_in[31];
    const int* g_idx = (const int*)d_in[32];

    const int N_ = in_sizes[27];             // 100000
    const int E_ = in_sizes[29];             // 1000000
    const int B_ = in_sizes[30];             // 20000
    const int* esrc = eidx;
    const int* edst = eidx + E_;

    // Workspace layout (floats)
    float* ws = (float*)d_ws;
    size_t off = 0;
    float* x      = ws + off; off += (size_t)N_ * 128;
    float* lu     = ws + off; off += (size_t)N_;
    float* q      = ws + off; off += (size_t)N_ * 128;
    float* k      = ws + off; off += (size_t)N_ * 128;
    float* v      = ws + off; off += (size_t)N_ * 128;
    float* skip   = ws + off; off += (size_t)N_ * 128;
    float* agg    = ws + off; off += (size_t)N_ * 128;  // becomes z
    float* h1     = ws + off; off += (size_t)N_ * 128;
    float* z2     = ws + off; off += (size_t)N_ * 64;
    float* logits = ws + off; off += (size_t)E_ * 2;
    float* wbuf   = ws + off; off += (size_t)E_ * 2;
    float* mmax   = ws + off; off += (size_t)N_ * 2;
    float* sbuf   = ws + off; off += (size_t)N_ * 2;
    float* bns    = ws + off; off += 128;
    float* bnq    = ws + off; off += 128;
    (void)ws_size; (void)n_in; (void)out_size;

    float* out_pos = (float*)d_out;
    float* out_neg = out_pos + B_;

    hipMemsetAsync(agg,  0, (size_t)N_ * 128 * sizeof(float), stream);
    hipMemsetAsync(sbuf, 0, (size_t)N_ * 2 * sizeof(float), stream);
    hipMemsetAsync(bns,  0, 256 * sizeof(float), stream);  // bns + bnq contiguous
    fill_neg_inf<<<(N_ * 2 + 255) / 256, 256, 0, stream>>>(mmax, N_ * 2);

    // 1. gather memory rows
    gather_kernel<<<((size_t)N_ * 128 + 255) / 256, 256, 0, stream>>>(
        memory_table, last_update, n_id, x, lu, N_);

    // 2. fused q/k/v/skip projections (x streamed once, A cached in registers)
    const int gemm_blocks = (N_ + 127) / 128;  // 8 waves x 16 rows per block
    node_gemm4_kernel<128, 128><<<gemm_blocks, 256, 0, stream>>>(
        x, Wq, bq, q, Wk, bk, k, Wv, bv, v, Wsk, bsk, skip, N_);

    // 3. attention passes (We staged in 64 KB dynamic LDS)
    const int edge_blocks = (E_ + 16 * EW - 1) / (16 * EW);
    const size_t we_bytes = 32 * 8 * 32 * sizeof(v2f);  // 64 KB
    edge_pass1<<<edge_blocks, 32 * EW, we_bytes, stream>>>(
        lu, graph_t, graph_msg, Wt, bt, We, q, k, esrc, edst, e_id, logits, mmax, E_);
    edge_pass2<<<(E_ * 2 + 255) / 256, 256, 0, stream>>>(logits, mmax, edst, wbuf, sbuf, E_);
    edge_pass3<<<edge_blocks, 32 * EW, we_bytes, stream>>>(
        lu, graph_t, graph_msg, Wt, bt, We, v, wbuf, sbuf, esrc, edst, e_id, agg, E_);

    // 4. skip connection, MLP with batch-norm
    add_skip_kernel<<<((size_t)N_ * 128 + 255) / 256, 256, 0, stream>>>(agg, skip, N_ * 128);
    node_gemm_kernel<128, 128><<<gemm_blocks, 256, 0, stream>>>(agg, W1, b1, h1, N_);
    bn_reduce<<<512, 128, 0, stream>>>(h1, bns, bnq, N_);
    bn_apply<<<((size_t)N_ * 128 + 255) / 256, 256, 0, stream>>>(h1, bns, bnq, bng, bnb, N_);
    node_gemm_kernel<128, 64><<<gemm_blocks, 256, 0, stream>>>(h1, W2, b2, z2, N_);

    // 5. link prediction (pos, neg)
    const int lp_blocks = (B_ + 16 * LW - 1) / (16 * LW);
    linkpred_kernel<<<lp_blocks, 32 * LW, 0, stream>>>(
        z2, s_idx, d_idx, Wls, bls, Wld, bld, Wlf, blf, out_pos, B_);
    linkpred_kernel<<<lp_blocks, 32 * LW, 0, stream>>>(
        z2, s_idx, g_idx, Wls, bls, Wld, bld, Wlf, blf, out_neg, B_);
}